// StructuralDiscriminator_77610059039275
// MI455X (gfx1250) — compile-verified
//
#include <hip/hip_runtime.h>

#define D_MODEL  768
#define RANK     64
#define N_LABELS 74
#define BATCH    8
#define SEQ      512
#define M_ROWS   (BATCH * SEQ)   // 4096
#define NPROJ    (2 * RANK)      // 128 (dist cols 0..63, depth cols 64..127)
#define KC       64              // K chunk staged per iteration in proj GEMM
#define NCHUNK   (D_MODEL / KC)  // 12
#define LDSPAD   68              // padded LDS row stride in floats (16B aligned, bank-conflict free)

typedef float v2f __attribute__((ext_vector_type(2)));
typedef float v4f __attribute__((ext_vector_type(4)));
typedef float v8f __attribute__((ext_vector_type(8)));

// ---------------------------------------------------------------------------
// CDNA5 async global->LDS 16-byte copy (ASYNCcnt-tracked, no VGPR data path).
// vdst = wave-relative LDS byte address, vaddr = 64-bit global address.
// ---------------------------------------------------------------------------
__device__ __forceinline__ void async_copy_b128(const float* gsrc, float* ldst) {
    unsigned lds_addr = (unsigned)(unsigned long long)(uintptr_t)ldst;
    asm volatile("global_load_async_to_lds_b128 %0, %1, off"
                 :: "v"(lds_addr), "v"(gsrc)
                 : "memory");
}
__device__ __forceinline__ void s_wait_asynccnt_0() {
    asm volatile("s_wait_asynccnt 0x0" ::: "memory");
}
__device__ __forceinline__ void s_wait_asynccnt_8() {  // allow next chunk's 8 in flight
    asm volatile("s_wait_asynccnt 0x8" ::: "memory");
}

// ---------------------------------------------------------------------------
// Kernel A: proj[4096x128] = enc[4096x768] @ [W_dist;W_depth]^T
// Block = 8 waves; Wcat K-chunks (128 cols x 64 K) double-buffered in LDS via
// async-to-LDS; A fragments register-batched; V_WMMA_F32_16X16X4_F32 math.
// ---------------------------------------------------------------------------
__global__ void __launch_bounds__(256) proj_wmma_kernel(
        const float* __restrict__ enc,
        const float* __restrict__ Wdist,
        const float* __restrict__ Wdepth,
        float* __restrict__ proj) {
    __shared__ float bsm[2][NPROJ * LDSPAD];   // 2 x 34.8 KB

    const int tid   = (int)threadIdx.x;
    const int wv    = tid >> 5;
    const int lane  = tid & 31;
    const int laneM = lane & 15;
    const int half  = lane >> 4;               // 0 -> K{0,1}, 1 -> K{2,3}
    const int row0  = ((int)blockIdx.x * 8 + wv) * 16;

    const float* aBase = enc + (size_t)(row0 + laneM) * D_MODEL + 2 * half;

    // Stage one 128x64 chunk of Wcat into LDS: 2048 B128 segments, 8/thread.
    auto stage = [&](int c, float* dst) {
#pragma unroll
        for (int i = 0; i < 8; ++i) {
            const int seg = tid + 256 * i;        // 0..2047
            const int n   = seg >> 4;             // 0..127 (output col)
            const int kk  = (seg & 15) * 4;       // 0..60
            const float* src =
                ((n < RANK) ? Wdist + (size_t)n * D_MODEL
                            : Wdepth + (size_t)(n - RANK) * D_MODEL) + c * KC + kk;
            async_copy_b128(src, dst + n * LDSPAD + kk);
        }
    };

    stage(0, bsm[0]);

    v8f acc[8] = {};
    for (int c = 0; c < NCHUNK; ++c) {
        stage((c + 1) % NCHUNK, bsm[(c + 1) & 1]);   // prefetch next chunk
        s_wait_asynccnt_8();                         // chunk c resident
        __syncthreads();

        // Batch A fragments for this chunk (16 x b64, single loadcnt wait).
        v2f afr[KC / 4];
#pragma unroll
        for (int s = 0; s < KC / 4; ++s)
            afr[s] = *(const v2f*)(aBase + c * KC + s * 4);

        const float* bs = bsm[c & 1];
#pragma unroll
        for (int s = 0; s < KC / 4; ++s) {
            v2f bf[8];
#pragma unroll
            for (int j = 0; j < 8; ++j)
                bf[j] = *(const v2f*)(bs + (j * 16 + laneM) * LDSPAD + s * 4 + 2 * half);
#pragma unroll
            for (int j = 0; j < 8; ++j)
                acc[j] = __builtin_amdgcn_wmma_f32_16x16x4_f32(
                    false, afr[s], false, bf[j], (short)0, acc[j], false, false);
        }
        __syncthreads();   // all waves done with bsm[c&1] before it is restaged
    }

    // C/D layout: VGPR p holds (M = p + 8*half, N = laneM)
#pragma unroll
    for (int j = 0; j < 8; ++j)
#pragma unroll
        for (int p = 0; p < 8; ++p)
            proj[(size_t)(row0 + p + 8 * half) * NPROJ + j * 16 + laneM] = acc[j][p];
}

// ---------------------------------------------------------------------------
// Kernel B: per-row squared norms (dist -> nd scratch, depth -> norms output).
// ---------------------------------------------------------------------------
__global__ void norms_kernel(const float* __restrict__ proj,
                             float* __restrict__ nd,
                             float* __restrict__ norms) {
    const int row  = (int)((blockIdx.x * blockDim.x + threadIdx.x) >> 5);
    const int lane = (int)(threadIdx.x & 31);
    const float* r = proj + (size_t)row * NPROJ;
    const v2f dv = *(const v2f*)(r + lane * 2);
    const v2f pv = *(const v2f*)(r + RANK + lane * 2);
    float sd = dv.x * dv.x + dv.y * dv.y;
    float sp = pv.x * pv.x + pv.y * pv.y;
#pragma unroll
    for (int off = 16; off; off >>= 1) {
        sd += __shfl_xor(sd, off, 32);
        sp += __shfl_xor(sp, off, 32);
    }
    if (lane == 0) { nd[row] = sd; norms[row] = sp; }
}

// ---------------------------------------------------------------------------
// Kernel C: per-batch Gram G = D @ D^T with D = dist projection [512x64].
// Block = one 16-row i-tile x full 512 j; the whole D_b (512x64 f32) is
// async-staged into LDS once and shared by all 8 waves (wave wv -> 64 j cols).
// Epilogue: sqd[b][i][j] = nd[i] + nd[j] - 2*G[i][j].
// ---------------------------------------------------------------------------
__global__ void __launch_bounds__(256) gram_dist_kernel(
        const float* __restrict__ proj,
        const float* __restrict__ nd,
        float* __restrict__ sqd) {
    __shared__ float dsm[SEQ * LDSPAD];        // 512*68*4 = 136 KB (WGP LDS is 320 KB)

    const int tid   = (int)threadIdx.x;
    const int wv    = tid >> 5;
    const int lane  = tid & 31;
    const int laneM = lane & 15;
    const int half  = lane >> 4;
    const int b     = (int)blockIdx.x >> 5;    // 0..7
    const int i0    = ((int)blockIdx.x & 31) * 16;
    const int j0    = wv * 64;

    // Stage D_b: 512 rows x 64 dist floats = 8192 B128 segments, 32/thread.
#pragma unroll
    for (int i = 0; i < 32; ++i) {
        const int seg = tid + 256 * i;          // 0..8191
        const int row = seg >> 4;               // 0..511
        const int kk  = (seg & 15) * 4;         // 0..60
        async_copy_b128(proj + (size_t)(b * SEQ + row) * NPROJ + kk,
                        dsm + row * LDSPAD + kk);
    }
    s_wait_asynccnt_0();
    __syncthreads();

    v8f acc[4] = {};
#pragma unroll
    for (int s = 0; s < RANK / 4; ++s) {
        const v2f a = *(const v2f*)(dsm + (i0 + laneM) * LDSPAD + s * 4 + 2 * half);
        v2f bf[4];
#pragma unroll
        for (int jj = 0; jj < 4; ++jj)
            bf[jj] = *(const v2f*)(dsm + (j0 + jj * 16 + laneM) * LDSPAD + s * 4 + 2 * half);
#pragma unroll
        for (int jj = 0; jj < 4; ++jj)
            acc[jj] = __builtin_amdgcn_wmma_f32_16x16x4_f32(
                false, a, false, bf[jj], (short)0, acc[jj], false, false);
    }

    const float* ndb = nd + b * SEQ;
    float ndi[8];
#pragma unroll
    for (int p = 0; p < 8; ++p) ndi[p] = ndb[i0 + p + 8 * half];

#pragma unroll
    for (int jj = 0; jj < 4; ++jj) {
        const int j     = j0 + jj * 16 + laneM;
        const float ndj = ndb[j];
        float* o = sqd + (size_t)b * SEQ * SEQ + j;
#pragma unroll
        for (int p = 0; p < 8; ++p) {
            const int i = i0 + p + 8 * half;
            o[(size_t)i * SEQ] = ndi[p] + ndj - 2.0f * acc[jj][p];
        }
    }
}

// ---------------------------------------------------------------------------
// Kernel D: LayerNorm (no affine) + 74x768 matvec + bias; one block per batch.
// ---------------------------------------------------------------------------
__global__ void adv_head_kernel(const float* __restrict__ sent,
                                const float* __restrict__ Wadv,
                                const float* __restrict__ badv,
                                float* __restrict__ out) {
    __shared__ float xs[D_MODEL];
    __shared__ float wred[8];
    const int b    = (int)blockIdx.x;
    const int tid  = (int)threadIdx.x;
    const int lane = tid & 31;
    const int wv   = tid >> 5;
    const float* s = sent + (size_t)b * D_MODEL;

    float sum = 0.f;
    for (int i = tid; i < D_MODEL; i += 256) { const float v = s[i]; xs[i] = v; sum += v; }
#pragma unroll
    for (int off = 16; off; off >>= 1) sum += __shfl_xor(sum, off, 32);
    if (lane == 0) wred[wv] = sum;
    __syncthreads();
    float total = 0.f;
    for (int w = 0; w < 8; ++w) total += wred[w];
    const float mean = total / (float)D_MODEL;
    __syncthreads();

    float vsum = 0.f;
    for (int i = tid; i < D_MODEL; i += 256) { const float d = xs[i] - mean; vsum += d * d; }
#pragma unroll
    for (int off = 16; off; off >>= 1) vsum += __shfl_xor(vsum, off, 32);
    if (lane == 0) wred[wv] = vsum;
    __syncthreads();
    float vtot = 0.f;
    for (int w = 0; w < 8; ++w) vtot += wred[w];
    const float rstd = rsqrtf(vtot / (float)D_MODEL + 1e-5f);
    __syncthreads();

    for (int i = tid; i < D_MODEL; i += 256) xs[i] = (xs[i] - mean) * rstd;
    __syncthreads();

    for (int lbl = wv; lbl < N_LABELS; lbl += 8) {
        const float* wr = Wadv + (size_t)lbl * D_MODEL;
        float acc = 0.f;
        for (int i = lane; i < D_MODEL; i += 32) acc += xs[i] * wr[i];
#pragma unroll
        for (int off = 16; off; off >>= 1) acc += __shfl_xor(acc, off, 32);
        if (lane == 0) out[b * N_LABELS + lbl] = acc + badv[lbl];
    }
}

// ---------------------------------------------------------------------------
extern "C" void kernel_launch(void* const* d_in, const int* in_sizes, int n_in,
                              void* d_out, int out_size, void* d_ws, size_t ws_size,
                              hipStream_t stream) {
    (void)in_sizes; (void)n_in; (void)out_size; (void)ws_size;

    const float* enc    = (const float*)d_in[0];  // [8,512,768]
    const float* sent   = (const float*)d_in[1];  // [8,1,768]
    const float* Wdist  = (const float*)d_in[2];  // [64,768]
    const float* Wdepth = (const float*)d_in[3];  // [64,768]
    const float* Wadv   = (const float*)d_in[4];  // [74,768]
    const float* badv   = (const float*)d_in[5];  // [74]

    float* out       = (float*)d_out;
    float* out_sq    = out;                                    // 8*512*512
    float* out_norms = out + (size_t)BATCH * SEQ * SEQ;        // 8*512
    float* out_adv   = out_norms + BATCH * SEQ;                // 8*74

    float* proj = (float*)d_ws;                                // 4096*128 f32
    float* nd   = proj + (size_t)M_ROWS * NPROJ;               // 4096 f32

    proj_wmma_kernel<<<M_ROWS / 16 / 8, 256, 0, stream>>>(enc, Wdist, Wdepth, proj);
    norms_kernel<<<M_ROWS / 8, 256, 0, stream>>>(proj, nd, out_norms);
    gram_dist_kernel<<<BATCH * 32, 256, 0, stream>>>(proj, nd, out_sq);
    adv_head_kernel<<<BATCH, 256, 0, stream>>>(sent, Wadv, badv, out_adv);
}